// FC_ddp2_68934225101180
// MI455X (gfx1250) — compile-verified
//
#include <hip/hip_runtime.h>
#include <hip/hip_bf16.h>

#define B_ 256
#define D_ 512
#define C_ 100000
#define NB 4096
#define SCALE_ 64.0f
#define MARGIN_ 0.4f

typedef float v2f __attribute__((ext_vector_type(2)));
typedef float v8f __attribute__((ext_vector_type(8)));

struct Sel {
    float lo;
    float hi;
    int k;
    unsigned th_bits;
    int topk_sum;
};

// ---------------------------------------------------------------------------
// K1: row-normalize embeddings -> emb_n [256,512]
// ---------------------------------------------------------------------------
__global__ __launch_bounds__(256) void k_norm_emb(const float* __restrict__ emb,
                                                  float* __restrict__ emb_n) {
    __shared__ float red[256];
    int b = blockIdx.x;
    const float* row = emb + (size_t)b * D_;
    float s = 0.f;
    for (int d = threadIdx.x; d < D_; d += 256) { float v = row[d]; s += v * v; }
    red[threadIdx.x] = s;
    __syncthreads();
    for (int st = 128; st > 0; st >>= 1) {
        if (threadIdx.x < st) red[threadIdx.x] += red[threadIdx.x + st];
        __syncthreads();
    }
    float inv = rsqrtf(red[0]);
    for (int d = threadIdx.x; d < D_; d += 256)
        emb_n[(size_t)b * D_ + d] = row[d] * inv;
}

// ---------------------------------------------------------------------------
// K-init: zero histogram, init selection state
// ---------------------------------------------------------------------------
__global__ void k_init(unsigned* __restrict__ hist, Sel* __restrict__ sel) {
    int i = blockIdx.x * blockDim.x + threadIdx.x;
    if (i < NB) hist[i] = 0u;
    if (i == 0) {
        sel->lo = -3.001f;
        sel->hi = 1.001f;
        sel->k = 1;
        sel->th_bits = 0u;
        sel->topk_sum = 0;
    }
}

// ---------------------------------------------------------------------------
// K3: FP32 WMMA GEMM  cos[b,c] = clip( inv_colnorm[c] * sum_d emb_n[b,d]*K[d,c] )
// One wave handles 4 row-tiles (64 rows) x 1 col-tile (16 cols).
// Column sum-of-squares is fused from the same B loads (kernel read ONCE).
// ---------------------------------------------------------------------------
__global__ __launch_bounds__(256) void k_gemm(const float* __restrict__ An,
                                              const float* __restrict__ Kw,
                                              float* __restrict__ cosm) {
    int wid  = (blockIdx.x * 256 + threadIdx.x) >> 5;   // 0..24999
    int lane = threadIdx.x & 31;
    int half = lane >> 4;           // 0 or 1
    int m    = lane & 15;           // 0..15
    int colBlk = wid >> 2;          // 0..6249
    int rg     = wid & 3;           // row group: rows rg*64 .. rg*64+63
    int col    = (colBlk << 4) + m;
    int rowBase = rg * 64;

    v8f acc0 = {}; v8f acc1 = {}; v8f acc2 = {}; v8f acc3 = {};
    float nrm = 0.f;
    const float* a0 = An + (size_t)(rowBase +  0 + m) * D_;
    const float* a1 = An + (size_t)(rowBase + 16 + m) * D_;
    const float* a2 = An + (size_t)(rowBase + 32 + m) * D_;
    const float* a3 = An + (size_t)(rowBase + 48 + m) * D_;

    for (int k = 0; k < D_; k += 4) {
        int kk = k + 2 * half;      // A: lane holds M=m, K=kk..kk+1 ; B: N=m, K=kk..kk+1
        float bx = __builtin_nontemporal_load(&Kw[(size_t)kk * C_ + col]);
        float by = __builtin_nontemporal_load(&Kw[(size_t)(kk + 1) * C_ + col]);
        nrm += bx * bx + by * by;
        v2f b; b.x = bx; b.y = by;
        v2f a;
        a.x = a0[kk]; a.y = a0[kk + 1];
        acc0 = __builtin_amdgcn_wmma_f32_16x16x4_f32(false, a, false, b, (short)0, acc0, false, false);
        a.x = a1[kk]; a.y = a1[kk + 1];
        acc1 = __builtin_amdgcn_wmma_f32_16x16x4_f32(false, a, false, b, (short)0, acc1, false, false);
        a.x = a2[kk]; a.y = a2[kk + 1];
        acc2 = __builtin_amdgcn_wmma_f32_16x16x4_f32(false, a, false, b, (short)0, acc2, false, false);
        a.x = a3[kk]; a.y = a3[kk + 1];
        acc3 = __builtin_amdgcn_wmma_f32_16x16x4_f32(false, a, false, b, (short)0, acc3, false, false);
    }
    // lanes L and L+16 covered complementary K halves of the same column
    nrm += __shfl_xor(nrm, 16, 32);
    float inv = rsqrtf(nrm);

    v8f accs[4] = {acc0, acc1, acc2, acc3};
    #pragma unroll
    for (int j = 0; j < 4; ++j) {
        #pragma unroll
        for (int v = 0; v < 8; ++v) {
            float val = accs[j][v] * inv;
            val = fminf(1.f, fmaxf(-1.f, val));
            int r = rowBase + j * 16 + v + 8 * half;   // D layout: row = vgpr + 8*half
            cosm[(size_t)r * C_ + col] = val;
        }
    }
}

// ---------------------------------------------------------------------------
// K4: per-row target value, argmax (first-max tie rule), top-k count
// ---------------------------------------------------------------------------
__global__ __launch_bounds__(256) void k_rowstats1(const float* __restrict__ cosm,
                                                   const int* __restrict__ label,
                                                   float* __restrict__ tgt,
                                                   int* __restrict__ amax,
                                                   Sel* __restrict__ sel) {
    __shared__ float svals[256];
    __shared__ int   sidx[256];
    __shared__ int   scnt[256];
    int b = blockIdx.x;
    int lab = label[b];
    const float* row = cosm + (size_t)b * C_;
    float t = row[lab];
    float bestv = -2.f; int besti = 0;
    int cnt = 0;
    for (int c = threadIdx.x; c < C_; c += 256) {
        float v = row[c];
        if (v > bestv) { bestv = v; besti = c; }
        if (c != lab && v > t) cnt++;
    }
    svals[threadIdx.x] = bestv; sidx[threadIdx.x] = besti; scnt[threadIdx.x] = cnt;
    __syncthreads();
    for (int s = 128; s > 0; s >>= 1) {
        if (threadIdx.x < s) {
            float v2 = svals[threadIdx.x + s]; int i2 = sidx[threadIdx.x + s];
            if (v2 > svals[threadIdx.x] ||
                (v2 == svals[threadIdx.x] && i2 < sidx[threadIdx.x])) {
                svals[threadIdx.x] = v2; sidx[threadIdx.x] = i2;
            }
            scnt[threadIdx.x] += scnt[threadIdx.x + s];
        }
        __syncthreads();
    }
    if (threadIdx.x == 0) {
        tgt[b] = t;
        amax[b] = sidx[0];
        atomicAdd(&sel->topk_sum, scnt[0]);
    }
}

// ---------------------------------------------------------------------------
// K5: far_rank = ceil(far * (B*(C-1) - topk_sum))  (f32 math, like the ref)
// ---------------------------------------------------------------------------
__global__ void k_set_k(Sel* __restrict__ sel) {
    float far = 1.0f / (float)(C_ - 1);
    float v = far * (float)(B_ * (C_ - 1) - sel->topk_sum);
    int fr = (int)ceilf(v);
    if (fr < 1) fr = 1;
    sel->k = fr;
}

// flat value f = tmp - 2*topk  (tmp = cos - 2*onehot)
__device__ __forceinline__ float flat_val(float v, int c, int lab, float t) {
    float tmp = (c == lab) ? v - 2.f : v;
    return (tmp > t) ? tmp - 2.f : tmp;
}

// ---------------------------------------------------------------------------
// K6: histogram of flat values in (lo, hi]
// ---------------------------------------------------------------------------
__global__ __launch_bounds__(256) void k_hist(const float* __restrict__ cosm,
                                              const int* __restrict__ label,
                                              const float* __restrict__ tgt,
                                              unsigned* __restrict__ hist,
                                              const Sel* __restrict__ sel) {
    __shared__ unsigned h[NB];
    for (int i = threadIdx.x; i < NB; i += 256) h[i] = 0u;
    __syncthreads();
    int b = blockIdx.x;
    float lo = sel->lo, hi = sel->hi;
    float scale = (float)NB / (hi - lo);
    int lab = label[b];
    float t = tgt[b];
    const float* row = cosm + (size_t)b * C_;
    for (int c = threadIdx.x; c < C_; c += 256) {
        float f = flat_val(row[c], c, lab, t);
        if (f > lo && f <= hi) {
            int bin = (int)((f - lo) * scale);
            bin = bin < 0 ? 0 : (bin > NB - 1 ? NB - 1 : bin);
            atomicAdd(&h[bin], 1u);
        }
    }
    __syncthreads();
    for (int i = threadIdx.x; i < NB; i += 256) {
        unsigned v = h[i];
        if (v) atomicAdd(&hist[i], v);
    }
}

// ---------------------------------------------------------------------------
// K7: narrow the selection window to the bin holding the k-th largest
// ---------------------------------------------------------------------------
__global__ void k_narrow(unsigned* __restrict__ hist, Sel* __restrict__ sel) {
    float lo = sel->lo, hi = sel->hi;
    int kk = sel->k;
    float w = (hi - lo) / (float)NB;
    long long cum = 0;
    int bin = -1;
    for (int i = NB - 1; i >= 0; --i) {
        long long c = (long long)hist[i];
        if (cum + c >= (long long)kk) { bin = i; kk -= (int)cum; break; }
        cum += c;
    }
    if (bin < 0) {  // defensive: take lowest nonempty bin
        for (int i = 0; i < NB; ++i)
            if (hist[i]) { bin = i; kk = (int)hist[i]; break; }
        if (bin < 0) { bin = 0; kk = 1; }
    }
    sel->lo = lo + w * (float)bin;
    sel->hi = lo + w * (float)(bin + 1);
    sel->k = kk;
    for (int i = 0; i < NB; ++i) hist[i] = 0u;
}

// ---------------------------------------------------------------------------
// K8: pick the exact threshold value in the final (sub-ulp) window
// ---------------------------------------------------------------------------
__global__ __launch_bounds__(256) void k_pick(const float* __restrict__ cosm,
                                              const int* __restrict__ label,
                                              const float* __restrict__ tgt,
                                              Sel* __restrict__ sel) {
    int b = blockIdx.x;
    float lo = sel->lo, hi = sel->hi;
    int lab = label[b];
    float t = tgt[b];
    const float* row = cosm + (size_t)b * C_;
    unsigned best = 0u;
    for (int c = threadIdx.x; c < C_; c += 256) {
        float f = flat_val(row[c], c, lab, t);
        if (f > lo && f <= hi) {
            unsigned bits = __float_as_uint(f);
            unsigned e = (bits & 0x80000000u) ? ~bits : (bits | 0x80000000u);
            best = best > e ? best : e;
        }
    }
    if (best) atomicMax(&sel->th_bits, best);
}

// ---------------------------------------------------------------------------
// K9: per-row negative stats -> margined target tgt_m
// ---------------------------------------------------------------------------
__global__ __launch_bounds__(256) void k_rowstats2(const float* __restrict__ cosm,
                                                   const int* __restrict__ label,
                                                   const float* __restrict__ tgt,
                                                   float* __restrict__ tgtm,
                                                   const Sel* __restrict__ sel) {
    __shared__ float ssum[256];
    __shared__ int   stimes[256];
    unsigned u = sel->th_bits;
    float th;
    if (u == 0u) th = 1e30f;
    else {
        unsigned bits = (u & 0x80000000u) ? (u & 0x7FFFFFFFu) : ~u;
        th = __uint_as_float(bits);
    }
    int b = blockIdx.x;
    int lab = label[b];
    float t = tgt[b];
    const float* row = cosm + (size_t)b * C_;
    float sum = 0.f; int times = 0;
    for (int c = threadIdx.x; c < C_; c += 256) {
        float v = row[c];
        float tmp = (c == lab) ? v - 2.f : v;
        bool topk = tmp > t;
        bool cond = (!topk) && (tmp > th);
        float neg = cond ? tmp : 0.f;
        float nsq = neg * neg;          // neg == neg_d in forward
        sum += nsq;
        if (nsq > 0.f) times++;
    }
    ssum[threadIdx.x] = sum; stimes[threadIdx.x] = times;
    __syncthreads();
    for (int s = 128; s > 0; s >>= 1) {
        if (threadIdx.x < s) {
            ssum[threadIdx.x]   += ssum[threadIdx.x + s];
            stimes[threadIdx.x] += stimes[threadIdx.x + s];
        }
        __syncthreads();
    }
    if (threadIdx.x == 0) {
        float tm = fmaxf((float)stimes[0], 1.f);
        float neg_mean = ssum[0] / tm;
        float tgm = t - MARGIN_;
        tgm = tgm - (1.f + t) * neg_mean;   // tgt_d == tgt numerically
        tgtm[b] = tgm;
    }
}

// ---------------------------------------------------------------------------
// K10: per-row streaming log-sum-exp of SCALE * (cos with label replaced)
// ---------------------------------------------------------------------------
__global__ __launch_bounds__(256) void k_lse(const float* __restrict__ cosm,
                                             const int* __restrict__ label,
                                             const float* __restrict__ tgtm,
                                             float* __restrict__ lse) {
    __shared__ float sm[256], ss[256];
    int b = blockIdx.x;
    int lab = label[b];
    float tg = tgtm[b];
    const float* row = cosm + (size_t)b * C_;
    float m = -1e30f, s = 0.f;
    for (int c = threadIdx.x; c < C_; c += 256) {
        float v = ((c == lab) ? tg : row[c]) * SCALE_;
        if (v > m) { s = s * expf(m - v) + 1.f; m = v; }
        else s += expf(v - m);
    }
    sm[threadIdx.x] = m; ss[threadIdx.x] = s;
    __syncthreads();
    for (int st = 128; st > 0; st >>= 1) {
        if (threadIdx.x < st) {
            float m1 = sm[threadIdx.x], s1 = ss[threadIdx.x];
            float m2 = sm[threadIdx.x + st], s2 = ss[threadIdx.x + st];
            float M = fmaxf(m1, m2);
            ss[threadIdx.x] = s1 * expf(m1 - M) + s2 * expf(m2 - M);
            sm[threadIdx.x] = M;
        }
        __syncthreads();
    }
    if (threadIdx.x == 0) lse[b] = sm[0] + logf(ss[0]);
}

// ---------------------------------------------------------------------------
// K11: final reduction -> d_out[0]=loss, d_out[1]=acc
// ---------------------------------------------------------------------------
__global__ __launch_bounds__(256) void k_final(const float* __restrict__ tgtm,
                                               const float* __restrict__ lse,
                                               const int* __restrict__ amax,
                                               const int* __restrict__ label,
                                               float* __restrict__ out) {
    __shared__ float sl[256], sa[256];
    int b = threadIdx.x;
    float l = -(tgtm[b] * SCALE_ - lse[b]);
    float a = (amax[b] == label[b]) ? 1.f : 0.f;
    sl[b] = l; sa[b] = a;
    __syncthreads();
    for (int s = 128; s > 0; s >>= 1) {
        if (b < s) { sl[b] += sl[b + s]; sa[b] += sa[b + s]; }
        __syncthreads();
    }
    if (b == 0) { out[0] = sl[0] / (float)B_; out[1] = sa[0] / (float)B_; }
}

// ---------------------------------------------------------------------------
extern "C" void kernel_launch(void* const* d_in, const int* in_sizes, int n_in,
                              void* d_out, int out_size, void* d_ws, size_t ws_size,
                              hipStream_t stream) {
    const float* d_emb   = (const float*)d_in[0];   // [256,512]
    const int*   d_label = (const int*)d_in[1];     // [256]
    const float* d_kern  = (const float*)d_in[2];   // [512,100000]
    float* out = (float*)d_out;                     // [loss, acc]

    char* ws = (char*)d_ws;
    float* emb_n = (float*)ws;                                  // 512 KB
    float* cosm  = (float*)(ws + (size_t)B_ * D_ * 4);          // 102.4 MB
    size_t stOff = (size_t)B_ * D_ * 4 + (size_t)B_ * C_ * 4;
    float*    tgt  = (float*)(ws + stOff);
    int*      amax = (int*)(ws + stOff + 1024);
    float*    tgtm = (float*)(ws + stOff + 2048);
    float*    lse  = (float*)(ws + stOff + 3072);
    unsigned* hist = (unsigned*)(ws + stOff + 4096);
    Sel*      sel  = (Sel*)(ws + stOff + 4096 + NB * 4);

    k_norm_emb<<<B_, 256, 0, stream>>>(d_emb, emb_n);
    k_init<<<(NB + 255) / 256, 256, 0, stream>>>(hist, sel);
    k_gemm<<<3125, 256, 0, stream>>>(emb_n, d_kern, cosm);      // 25000 waves
    k_rowstats1<<<B_, 256, 0, stream>>>(cosm, d_label, tgt, amax, sel);
    k_set_k<<<1, 1, 0, stream>>>(sel);
    for (int r = 0; r < 3; ++r) {
        k_hist<<<B_, 256, 0, stream>>>(cosm, d_label, tgt, hist, sel);
        k_narrow<<<1, 1, 0, stream>>>(hist, sel);
    }
    k_pick<<<B_, 256, 0, stream>>>(cosm, d_label, tgt, sel);
    k_rowstats2<<<B_, 256, 0, stream>>>(cosm, d_label, tgt, tgtm, sel);
    k_lse<<<B_, 256, 0, stream>>>(cosm, d_label, tgtm, lse);
    k_final<<<1, 256, 0, stream>>>(tgtm, lse, amax, d_label, out);
}